// GlobalStyleTokenLayer_1065151889864
// MI455X (gfx1250) — compile-verified
//
#include <hip/hip_runtime.h>
#include <hip/hip_bf16.h>

// GST layer: ref_emb (32,2048,96) f32, tokens (10,96), Wq (96,96), Wk (96,96), Wv (4,96)
// out: (32,2048,96) f32.
//
// Strategy for MI455X (gfx1250, wave32):
//  - q = tanh(ref_emb @ Wq^T) via V_WMMA_F32_16X16X4_F32 (full fp32, matches reference).
//  - 69M tanh via hardware v_tanh_f32 (CDNA5 TRANS op).
//  - HBM floor ~50MB -> ~2.2us; Wq/Wv/tokens/k all stay in LDS or cache.

typedef float v2f __attribute__((ext_vector_type(2)));
typedef float v8f __attribute__((ext_vector_type(8)));

#if defined(__AMDGCN__) && __has_builtin(__builtin_amdgcn_tanhf)
__device__ __forceinline__ float tanh_fast(float x) {
  return __builtin_amdgcn_tanhf(x);
}
#else
__device__ __forceinline__ float tanh_fast(float x) {
  float y;
  asm volatile("v_tanh_f32 %0, %1\n\tv_nop" : "=v"(y) : "v"(x));
  return y;
}
#endif

#define DDIM 96
#define NTOK 10
#define NHEAD 4
#define TOTAL_ROWS (32 * 2048)

// ---------------- kernel 1: k = tanh(tokens @ Wk^T), 10x96 -> workspace ------
__global__ void gst_k_kernel(const float* __restrict__ tokens,
                             const float* __restrict__ Wk,
                             float* __restrict__ kout) {
  int idx = blockIdx.x * blockDim.x + threadIdx.x;
  if (idx >= NTOK * DDIM) return;
  int n = idx / DDIM;
  int d = idx % DDIM;
  float s = 0.f;
#pragma unroll 8
  for (int j = 0; j < DDIM; ++j)
    s = fmaf(tokens[n * DDIM + j], Wk[d * DDIM + j], s);
  kout[idx] = tanh_fast(s);
}

// ---------------- kernel 2: main fused GST -----------------------------------
// 128 threads = 4 waves; each wave owns a 16-row tile of ref_emb.
__global__ __launch_bounds__(128) void gst_main_kernel(
    const float* __restrict__ ref_emb,  // (65536, 96)
    const float* __restrict__ tokens,   // (10, 96)
    const float* __restrict__ Wq,       // (96, 96)
    const float* __restrict__ Wv,       // (4, 96)
    const float* __restrict__ kmat,     // (10, 96) precomputed tanh(tok@Wk^T)
    float* __restrict__ out) {          // (65536, 96)
  __shared__ float Wv_s[NHEAD * DDIM];     //  1536 B
  __shared__ float tok_s[NTOK * DDIM];     //  3840 B
  __shared__ float k_s[NTOK * DDIM];       //  3840 B
  __shared__ float Qs[4][16 * 97];         // 24832 B (stride 97: conflict-free)

  const int tid = threadIdx.x;
  for (int i = tid; i < NHEAD * DDIM; i += 128) Wv_s[i] = Wv[i];
  for (int i = tid; i < NTOK * DDIM; i += 128) tok_s[i] = tokens[i];
  for (int i = tid; i < NTOK * DDIM; i += 128) k_s[i] = kmat[i];
  __syncthreads();

  const int wave = tid >> 5;
  const int lane = tid & 31;
  const int row_base = (blockIdx.x * 4 + wave) * 16;

  // ---- fp32 WMMA 16x16x4: q_tile(16x96) = X(16x96) @ Wq^T ----
  // A (16x4) layout: lanes 0-15 -> M=0..15 ; VGPR0 holds K (lanes 0-15) and
  // K+2 (lanes 16-31), VGPR1 holds K+1 / K+3. B (4x16): lanes index N,
  // lane-half selects K / K+2 symmetrically.
  const int lrow = lane & 15;              // A row / B column-in-tile
  const int koff = (lane >> 4) << 1;       // 0 for lanes 0-15, 2 for 16-31
  const float* X = ref_emb + (size_t)row_base * DDIM;

  v8f acc[6] = {};
  for (int kk = 0; kk < DDIM; kk += 4) {
    const float* ap = X + (size_t)lrow * DDIM + kk + koff;
    v2f a;
    a.x = ap[0];
    a.y = ap[1];
#pragma unroll
    for (int nt = 0; nt < 6; ++nt) {
      // B[k][n] = (Wq^T)[k][n] = Wq[n][k]
      const float* bp = Wq + (size_t)(nt * 16 + lrow) * DDIM + kk + koff;
      v2f b;
      b.x = bp[0];
      b.y = bp[1];
      acc[nt] = __builtin_amdgcn_wmma_f32_16x16x4_f32(
          /*neg_a=*/false, a, /*neg_b=*/false, b,
          /*c_mod=*/(short)0, acc[nt], /*reuse_a=*/false, /*reuse_b=*/false);
    }
  }

  // tanh in-register, park q tile in LDS.
  // C/D layout: VGPR j -> M=j (lanes 0-15) / M=j+8 (lanes 16-31); N=lane&15.
  const int m_off = (lane >> 4) * 8;
#pragma unroll
  for (int nt = 0; nt < 6; ++nt) {
#pragma unroll
    for (int j = 0; j < 8; ++j) {
      Qs[wave][(j + m_off) * 97 + nt * 16 + lrow] = tanh_fast(acc[nt][j]);
    }
  }
  // Only this wave reads its own Qs slab; compiler orders DS store->load.

  // ---- attention phase: 2 lanes per row, each covers 48 of 96 dims ----
  const int row = lane >> 1;
  const int half = lane & 1;
  const int d0 = half * 48;
  const float* q = &Qs[wave][row * 97 + d0];

  float p[NTOK][NHEAD];
#pragma unroll
  for (int n = 0; n < NTOK; ++n) {
    float ph0 = 0.f, ph1 = 0.f, ph2 = 0.f, ph3 = 0.f;
    const float* kn = &k_s[n * DDIM + d0];
#pragma unroll 4
    for (int d = 0; d < 48; ++d) {
      float s = tanh_fast(q[d] + kn[d]);
      ph0 = fmaf(s, Wv_s[0 * DDIM + d0 + d], ph0);
      ph1 = fmaf(s, Wv_s[1 * DDIM + d0 + d], ph1);
      ph2 = fmaf(s, Wv_s[2 * DDIM + d0 + d], ph2);
      ph3 = fmaf(s, Wv_s[3 * DDIM + d0 + d], ph3);
    }
    // pair-reduce halves of the row (lanes 2r and 2r+1)
    p[n][0] = ph0 + __shfl_xor(ph0, 1, 32);
    p[n][1] = ph1 + __shfl_xor(ph1, 1, 32);
    p[n][2] = ph2 + __shfl_xor(ph2, 1, 32);
    p[n][3] = ph3 + __shfl_xor(ph3, 1, 32);
  }

  // softmax over tokens per head; fold mean-over-heads into per-token weight
  float w[NTOK];
#pragma unroll
  for (int n = 0; n < NTOK; ++n) w[n] = 0.f;
#pragma unroll
  for (int h = 0; h < NHEAD; ++h) {
    float mx = p[0][h];
#pragma unroll
    for (int n = 1; n < NTOK; ++n) mx = fmaxf(mx, p[n][h]);
    float sum = 0.f;
#pragma unroll
    for (int n = 0; n < NTOK; ++n) {
      float e = __expf(p[n][h] - mx);
      p[n][h] = e;
      sum += e;
    }
    float inv = 1.0f / sum;
#pragma unroll
    for (int n = 0; n < NTOK; ++n) w[n] = fmaf(p[n][h], inv, w[n]);
  }
#pragma unroll
  for (int n = 0; n < NTOK; ++n) w[n] *= 0.25f;

  // style[d] = sum_n w[n] * tokens[n][d]; 48 contiguous floats per lane
  float* outp = out + (size_t)(row_base + row) * DDIM + d0;
#pragma unroll 4
  for (int d = 0; d < 48; ++d) {
    float acc2 = 0.f;
#pragma unroll
    for (int n = 0; n < NTOK; ++n)
      acc2 = fmaf(w[n], tok_s[n * DDIM + d0 + d], acc2);
    outp[d] = acc2;
  }
}

extern "C" void kernel_launch(void* const* d_in, const int* in_sizes, int n_in,
                              void* d_out, int out_size, void* d_ws, size_t ws_size,
                              hipStream_t stream) {
  const float* ref_emb = (const float*)d_in[0];
  const float* tokens  = (const float*)d_in[1];
  const float* Wq      = (const float*)d_in[2];
  const float* Wk      = (const float*)d_in[3];
  const float* Wv      = (const float*)d_in[4];
  float* out = (float*)d_out;
  float* kws = (float*)d_ws;  // 10*96 floats

  gst_k_kernel<<<4, 256, 0, stream>>>(tokens, Wk, kws);

  // 65536 rows / 16 per wave / 4 waves per block = 1024 blocks
  gst_main_kernel<<<1024, 128, 0, stream>>>(ref_emb, tokens, Wq, Wv, kws, out);
}